// ComSim_13597866459340
// MI455X (gfx1250) — compile-verified
//
#include <hip/hip_runtime.h>

typedef __attribute__((ext_vector_type(2))) float v2f;
typedef __attribute__((ext_vector_type(8))) float v8f;

#define DWORDS_D 300
#define S_CATS 256
#define A_CATS 256
#define WPC 8
#define NROWS 2048            // S*W == A*W
#define COS_EPS 1e-8f

// ---------------------------------------------------------------------------
// Kernel A: gather embedding rows for sen/ann word indices, compute per-row
// L2 norm, and (sen only) the "row == none embedding" mask (faithful value
// compare against emb[none_idx], like the torch.equal branch).
// grid.x = 2*NROWS (first half sen rows, second half ann rows), block = 128.
// ---------------------------------------------------------------------------
__global__ void ComSim_gather_rows(const float* __restrict__ emb,
                                   const int* __restrict__ senCats,
                                   const int* __restrict__ annCats,
                                   const int* __restrict__ noneIdxPtr,
                                   float* __restrict__ senE,
                                   float* __restrict__ annE,
                                   float* __restrict__ ns,
                                   float* __restrict__ na,
                                   float* __restrict__ noneMask)
{
    __shared__ float redF[128];
    __shared__ int   redI[128];

    const int bid   = blockIdx.x;
    const int tid   = threadIdx.x;
    const bool isSen = bid < NROWS;
    const int row   = isSen ? bid : bid - NROWS;
    const int idx   = isSen ? senCats[row] : annCats[row];
    const int nIdx  = noneIdxPtr[0];

    const float* __restrict__ src  = emb + (long long)idx  * DWORDS_D;
    const float* __restrict__ nsrc = emb + (long long)nIdx * DWORDS_D;
    float* __restrict__ dst = (isSen ? senE : annE) + (long long)row * DWORDS_D;

    float ss = 0.0f;
    int eq = 1;
    for (int d = tid; d < DWORDS_D; d += 128) {
        float v = src[d];
        dst[d] = v;
        ss += v * v;
        if (isSen) eq &= (v == nsrc[d]) ? 1 : 0;
    }
    redF[tid] = ss;
    redI[tid] = eq;
    __syncthreads();
    for (int off = 64; off > 0; off >>= 1) {
        if (tid < off) {
            redF[tid] += redF[tid + off];
            redI[tid] &= redI[tid + off];
        }
        __syncthreads();
    }
    if (tid == 0) {
        float nrm = sqrtf(redF[0]);
        if (isSen) {
            ns[row] = nrm;
            noneMask[row] = redI[0] ? 1.0f : 0.0f;
        } else {
            na[row] = nrm;
        }
    }
}

// ---------------------------------------------------------------------------
// Kernel B: sen_wd_emb[s, :] = sum over the 8 gathered rows of sentence s.
// grid.x = 256, block = 128.
// ---------------------------------------------------------------------------
__global__ void ComSim_sen_word_sum(const float* __restrict__ senE,
                                    float* __restrict__ out)
{
    const int s = blockIdx.x;
    const int tid = threadIdx.x;
    for (int d = tid; d < DWORDS_D; d += 128) {
        float acc = 0.0f;
#pragma unroll
        for (int w = 0; w < WPC; ++w)
            acc += senE[((long long)(s * WPC + w)) * DWORDS_D + d];
        out[(long long)s * DWORDS_D + d] = acc;
    }
}

// ---------------------------------------------------------------------------
// Kernel C: one wave computes a 16x16 tile of the [2048 x 2048] dot matrix
// via V_WMMA_F32_16X16X4_F32 (K=300 in steps of 4), applies cosine
// normalization + none mask, then 4 lanes run the faithful order-dependent
// fold (p = w*8+v; cur = x if x>=cur || x<0) for the 2x2 (s,a) cells the
// tile covers. grid = (128, 128), block = 32 (one wave, EXEC all-ones for
// the WMMA section as the ISA requires).
// ---------------------------------------------------------------------------
__global__ void ComSim_cos_tile_wmma(const float* __restrict__ senE,
                                     const float* __restrict__ annE,
                                     const float* __restrict__ ns,
                                     const float* __restrict__ na,
                                     const float* __restrict__ noneMask,
                                     float* __restrict__ maxSim)
{
    __shared__ float tile[16 * 16];

    const int lane = threadIdx.x;        // 0..31
    const int half = lane >> 4;          // 0 or 1
    const int l16  = lane & 15;
    const int row0 = blockIdx.y * 16;    // sen word-row base
    const int col0 = blockIdx.x * 16;    // ann word-row base

    // A fragment: A[M=l16][K = k + 2*half + {0,1}]  (16x4 f32 layout)
    // B fragment: B[K][N=l16] with B[k][n] = annE[col0+n][k] -> same loads.
    const float* __restrict__ aRow = senE + (long long)(row0 + l16) * DWORDS_D + 2 * half;
    const float* __restrict__ bRow = annE + (long long)(col0 + l16) * DWORDS_D + 2 * half;

    v8f c = {};
#pragma unroll 5
    for (int k = 0; k < DWORDS_D; k += 4) {
        v2f a = *(const v2f*)(aRow + k);
        v2f b = *(const v2f*)(bRow + k);
        c = __builtin_amdgcn_wmma_f32_16x16x4_f32(
                /*neg_a=*/false, a, /*neg_b=*/false, b,
                /*c_mod=*/(short)0, c, /*reuse_a=*/false, /*reuse_b=*/false);
    }

    // D layout: element (M = r + 8*half, N = l16) lives in c[r].
    const float naV = na[col0 + l16];
#pragma unroll
    for (int r = 0; r < 8; ++r) {
        const int m = r + 8 * half;
        const float nsV   = ns[row0 + m];
        const float denom = fmaxf(nsV * naV, COS_EPS);
        float cosv = c[r] / denom;
        if (noneMask[row0 + m] != 0.0f) cosv = 0.0f;
        tile[m * 16 + l16] = cosv;
    }
    __syncthreads();

    if (lane < 4) {
        const int si = lane >> 1;        // which of the 2 sentences in tile
        const int ai = lane & 1;         // which of the 2 annotations
        float cur = 0.0f;
        for (int w = 0; w < WPC; ++w) {
            for (int v = 0; v < WPC; ++v) {
                float x = tile[(si * 8 + w) * 16 + (ai * 8 + v)];
                cur = ((x >= cur) || (x < 0.0f)) ? x : cur;   // faithful rule
            }
        }
        const int sIdx = blockIdx.y * 2 + si;
        const int aIdx = blockIdx.x * 2 + ai;
        maxSim[sIdx * A_CATS + aIdx] = cur;
    }
}

// ---------------------------------------------------------------------------
// Launcher
// ---------------------------------------------------------------------------
extern "C" void kernel_launch(void* const* d_in, const int* in_sizes, int n_in,
                              void* d_out, int out_size, void* d_ws, size_t ws_size,
                              hipStream_t stream) {
    const float* emb      = (const float*)d_in[0];   // [100000, 300]
    const int*   senCats  = (const int*)d_in[1];     // [256, 8]
    const int*   annCats  = (const int*)d_in[2];     // [256, 8]
    const int*   noneIdx  = (const int*)d_in[3];     // [1]

    float* out    = (float*)d_out;
    float* maxSim = out;                         // [256*256]
    float* senWd  = out + S_CATS * A_CATS;       // [256*300]

    // Workspace layout (bytes):
    //   senE     @ 0        : 2048*300*4 = 2,457,600
    //   annE     @ 2457600  : 2,457,600
    //   ns       @ 4915200  : 8,192
    //   na       @ 4923392  : 8,192
    //   noneMask @ 4931584  : 8,192      (total ~4.94 MB)
    char* ws = (char*)d_ws;
    float* senE     = (float*)(ws + 0);
    float* annE     = (float*)(ws + 2457600);
    float* ns       = (float*)(ws + 4915200);
    float* na       = (float*)(ws + 4923392);
    float* noneMask = (float*)(ws + 4931584);

    ComSim_gather_rows<<<dim3(2 * NROWS), dim3(128), 0, stream>>>(
        emb, senCats, annCats, noneIdx, senE, annE, ns, na, noneMask);

    ComSim_sen_word_sum<<<dim3(S_CATS), dim3(128), 0, stream>>>(senE, senWd);

    ComSim_cos_tile_wmma<<<dim3(128, 128), dim3(32), 0, stream>>>(
        senE, annE, ns, na, noneMask, maxSim);
}